// DetectionHead_90400471646691
// MI455X (gfx1250) — compile-verified
//
#include <hip/hip_runtime.h>
#include <stdint.h>

// DetectionHead headforward stencil for MI455X (gfx1250).
// Memory-bound (≈268 MB traffic → ~11.5 us at 23.3 TB/s). Strategy:
// async global->LDS tile staging (CDNA5 ASYNCcnt path) + LDS-fed VALU max tree,
// float4 coalesced stores.

#define EPS_F     0.01f
#define IMG_H     2048
#define IMG_W     2048
#define TILE_W    128
#define TILE_H    16
#define LSTRIDE   (TILE_W + 8)          // 136 floats per LDS row (544 B, 16B-aligned)
#define NROWS     (TILE_H + 2)          // 18 rows incl. halo
#define NC4       (LSTRIDE / 4)         // 34 float4 slots per row
#define NSLOT     (NROWS * NC4)         // 612 slots per block

__device__ __forceinline__ void async_load_f4_to_lds(const float* gptr, float* lptr) {
    // VDST vgpr = LDS byte address (low 32 bits of the generic/shared pointer),
    // VADDR vgpr pair = 64-bit global address. Tracked by ASYNCcnt.
    uint32_t lds_addr = (uint32_t)(uintptr_t)lptr;
    asm volatile("global_load_async_to_lds_b128 %0, %1, off"
                 :: "v"(lds_addr), "v"(gptr) : "memory");
}

__device__ __forceinline__ void wait_asynccnt0() {
#if defined(__has_builtin) && __has_builtin(__builtin_amdgcn_s_wait_asynccnt)
    __builtin_amdgcn_s_wait_asynccnt(0);
#else
    asm volatile("s_wait_asynccnt 0" ::: "memory");
#endif
}

__global__ __launch_bounds__(256)
void detection_head_kernel(const float* __restrict__ x, float* __restrict__ out) {
    __shared__ __align__(16) float tile[NROWS * LSTRIDE];

    const int tid = threadIdx.x;
    const int c0  = blockIdx.x * TILE_W;   // tile origin column
    const int r0  = blockIdx.y * TILE_H;   // tile origin row
    const long plane = (long)blockIdx.z * IMG_H * IMG_W;
    const float* __restrict__ xb = x + plane;
    float* __restrict__ ob = out + plane;

    // ---- Stage (TILE_H+2) x (TILE_W+8) floats into LDS; zero-fill OOB halo ----
    // LDS row 0 maps to global row r0-1; LDS col 0 maps to global col c0-4.
    #pragma unroll
    for (int s = tid; s < NSLOT; s += 256) {
        const int row = s / NC4;
        const int c4  = s - row * NC4;
        const int gr  = r0 - 1 + row;
        const int gc  = c0 - 4 + 4 * c4;
        float* lp = &tile[row * LSTRIDE + 4 * c4];
        if ((unsigned)gr < (unsigned)IMG_H && gc >= 0 && gc <= (IMG_W - 4)) {
            async_load_f4_to_lds(xb + (long)gr * IMG_W + gc, lp);
        } else {
            float4 z = make_float4(0.f, 0.f, 0.f, 0.f);
            *(float4*)lp = z;   // ds_store_b128, drained by __syncthreads
        }
    }
    wait_asynccnt0();
    __syncthreads();

    // ---- Compute: each thread emits two float4 outputs (rows row0 and row0+8) ----
    const int col4 = tid & 31;   // 0..31 -> output cols c0 + 4*col4 .. +3
    const int row0 = tid >> 5;   // 0..7

    #pragma unroll
    for (int rr = 0; rr < 2; ++rr) {
        const int row = row0 + rr * 8;            // output row within tile, 0..15
        const int lc  = 4 + 4 * col4;             // LDS float index of output col c0+4*col4
        const float* pt = &tile[(row + 0) * LSTRIDE + lc];  // global row r0+row-1
        const float* pm = &tile[(row + 1) * LSTRIDE + lc];  // global row r0+row
        const float* pb = &tile[(row + 2) * LSTRIDE + lc];  // global row r0+row+1

        const float4 t4 = *(const float4*)pt;
        const float4 m4 = *(const float4*)pm;
        const float4 b4 = *(const float4*)pb;
        const float tv[6] = { pt[-1], t4.x, t4.y, t4.z, t4.w, pt[4] };
        const float mv[6] = { pm[-1], m4.x, m4.y, m4.z, m4.w, pm[4] };
        const float bv[6] = { pb[-1], b4.x, b4.y, b4.z, b4.w, pb[4] };

        float o[4];
        #pragma unroll
        for (int j = 0; j < 4; ++j) {
            // max over the 8 neighbors (center excluded); OOB lanes are 0 in LDS.
            float m = fmaxf(fmaxf(fmaxf(tv[j], tv[j + 1]), fmaxf(tv[j + 2], bv[j])),
                            fmaxf(fmaxf(bv[j + 1], bv[j + 2]), fmaxf(mv[j], mv[j + 2])));
            // relu monotone: max_i relu(a_i - eps) == relu(max_i a_i - eps);
            // zero padding is reproduced by the 0s staged into LDS.
            const float xm = fmaxf(m - EPS_F, 0.f);
            const float xc = mv[j + 1];
            const float xp = fmaxf(xc - EPS_F, 0.f);
            o[j] = (xc > xm) ? xp : 0.f;
        }
        float4 ov = make_float4(o[0], o[1], o[2], o[3]);
        *(float4*)&ob[(long)(r0 + row) * IMG_W + (c0 + 4 * col4)] = ov;  // coalesced b128
    }
}

extern "C" void kernel_launch(void* const* d_in, const int* in_sizes, int n_in,
                              void* d_out, int out_size, void* d_ws, size_t ws_size,
                              hipStream_t stream) {
    const float* x = (const float*)d_in[0];
    float* out = (float*)d_out;
    const int batch = in_sizes[0] / (IMG_H * IMG_W);   // 8 for the reference shapes

    dim3 grid(IMG_W / TILE_W, IMG_H / TILE_H, batch);  // 16 x 128 x 8 = 16384 blocks
    dim3 block(256);                                   // 8 wave32s per block
    hipLaunchKernelGGL(detection_head_kernel, grid, block, 0, stream, x, out);
}